// CRFLayer_1872605741241
// MI455X (gfx1250) — compile-verified
//
#include <hip/hip_runtime.h>
#include <hip/hip_bf16.h>

#define B_ 64
#define S_ 512
#define H_ 1024
#define T_ 48

typedef float v2f __attribute__((ext_vector_type(2)));
typedef float v8f __attribute__((ext_vector_type(8)));
typedef unsigned int u32x4 __attribute__((ext_vector_type(4)));
typedef int i32x4 __attribute__((ext_vector_type(4)));
typedef int i32x8 __attribute__((ext_vector_type(8)));

#if defined(__has_builtin)
#if __has_builtin(__builtin_amdgcn_tensor_load_to_lds) && __has_builtin(__builtin_amdgcn_s_wait_tensorcnt)
#define HAVE_TDM 1
#endif
#endif
#ifndef HAVE_TDM
#define HAVE_TDM 0
#endif

#if HAVE_TDM
// ---------------------------------------------------------------------------
// TDM descriptor (D#) for one 128-row x 48-col fp32 chunk of W -> LDS.
// Group0: count=1 | lds_addr | 57-bit global addr | type=2 ("image").
// Group1: data_size=4B, tensor_dim0=48, tensor_dim1=1024, tile=48x128,
//         tensor_dim0_stride=48.  Groups 2/3 unused (zero) for a 2D tile.
// ---------------------------------------------------------------------------
__device__ __forceinline__ void tdm_load_w_chunk(const float* gsrc, unsigned lds_addr) {
  const unsigned long long ga = (unsigned long long)(size_t)gsrc;
  u32x4 g0;
  g0[0] = 1u;                                           // count=1, user mode
  g0[1] = lds_addr;                                     // LDS byte address
  g0[2] = (unsigned)(ga & 0xFFFFFFFFu);                 // global addr [31:0]
  g0[3] = (unsigned)((ga >> 32) & 0x01FFFFFFu)          // global addr [56:32]
        | (2u << 30);                                   // type = 2
  i32x8 g1;
  g1[0] = 0x00020000;                 // workgroup_mask=0, data_size=2 (4B)
  g1[1] = (int)(48u << 16);           // tensor_dim0 = 48 (bits 79:48, low16)
  g1[2] = (int)(1024u << 16);         // dim0 hi=0 | tensor_dim1 = 1024 (low16)
  g1[3] = (int)(48u << 16);           // dim1 hi=0 | tile_dim0 = 48
  g1[4] = 128;                        // tile_dim1 = 128, tile_dim2 = 0
  g1[5] = 48;                         // tensor_dim0_stride = 48 (low32)
  g1[6] = 0;                          // stride0 hi | stride1 low16
  g1[7] = 0;                          // stride1 hi
  const i32x4 z4 = {0, 0, 0, 0};
#if __clang_major__ >= 23
  const i32x8 z8 = {0, 0, 0, 0, 0, 0, 0, 0};
  __builtin_amdgcn_tensor_load_to_lds(g0, g1, z4, z4, z8, 0);
#else
  __builtin_amdgcn_tensor_load_to_lds(g0, g1, z4, z4, 0);
#endif
}
#endif

// ---------------------------------------------------------------------------
// Kernel 1: features[M = B*S, T] = x[M, H] @ W[H, T] + b  (fp32 WMMA 16x16x4)
// Each wave: one 16-row M tile x all 3 N tiles.  W double-buffered in LDS via
// the Tensor Data Mover (wave 0 issues the DMA; TENSORcnt + barrier sync).
// ---------------------------------------------------------------------------
__global__ __launch_bounds__(256)
void gemm_features_kernel(const float* __restrict__ x,
                          const float* __restrict__ W,
                          const float* __restrict__ bias,
                          float* __restrict__ feat) {
  __shared__ float sW[2][128 * T_];           // 2 x 24 KB K-chunks of W

  const int tid  = threadIdx.x;
  const int wave = tid >> 5;
  const int lane = tid & 31;
  const int lm   = lane & 15;                 // row (A) / col (B,D) in tile
  const int lh   = lane >> 4;                 // K-half select for A/B
  const int row0 = (blockIdx.x * 8 + wave) * 16;

  v8f acc0 = {}; v8f acc1 = {}; v8f acc2 = {};

  const float* __restrict__ xrow = x + (size_t)(row0 + lm) * H_;

  // ---- prologue: fetch chunk 0 --------------------------------------------
#if HAVE_TDM
  if (wave == 0) {
    tdm_load_w_chunk(W, (unsigned)(size_t)(void*)&sW[0][0]);
    __builtin_amdgcn_s_wait_tensorcnt(0);
  }
#else
  for (int idx = tid; idx < 128 * T_; idx += 256)
    sW[0][idx] = W[idx];
#endif
  __syncthreads();

  for (int c = 0; c < H_ / 128; ++c) {
    const int cur = c & 1;
    const int kc  = c * 128;

    // ---- overlap: start DMA of next chunk into the other buffer ----------
    if (c + 1 < H_ / 128) {
#if HAVE_TDM
      if (wave == 0)
        tdm_load_w_chunk(W + (size_t)(kc + 128) * T_,
                         (unsigned)(size_t)(void*)&sW[1 - cur][0]);
#else
      for (int idx = tid; idx < 128 * T_; idx += 256)
        sW[1 - cur][idx] = W[(kc + 128) * T_ + idx];
#endif
      __builtin_prefetch(xrow + kc + 128 + 2 * lane, 0, 0);
    }

    const float* sWc = &sW[cur][0];

    // ---- compute on current buffer: 96 WMMAs ------------------------------
    for (int kk = 0; kk < 128; kk += 16) {
      // batch 4 independent A loads (16 K-values) before the 12 WMMAs
      const v2f a0 = *(const v2f*)(xrow + kc + kk +      2 * lh);
      const v2f a1 = *(const v2f*)(xrow + kc + kk +  4 + 2 * lh);
      const v2f a2 = *(const v2f*)(xrow + kc + kk +  8 + 2 * lh);
      const v2f a3 = *(const v2f*)(xrow + kc + kk + 12 + 2 * lh);

      auto wmma3 = [&](v2f a, int kb) {
        const float* wr0 = &sWc[(kb + 2 * lh) * T_];
        const float* wr1 = wr0 + T_;
        v2f b0; b0.x = wr0[lm];      b0.y = wr1[lm];
        v2f b1; b1.x = wr0[16 + lm]; b1.y = wr1[16 + lm];
        v2f b2; b2.x = wr0[32 + lm]; b2.y = wr1[32 + lm];
        acc0 = __builtin_amdgcn_wmma_f32_16x16x4_f32(false, a, false, b0, (short)0, acc0, false, false);
        acc1 = __builtin_amdgcn_wmma_f32_16x16x4_f32(false, a, false, b1, (short)0, acc1, false, false);
        acc2 = __builtin_amdgcn_wmma_f32_16x16x4_f32(false, a, false, b2, (short)0, acc2, false, false);
      };
      wmma3(a0, kk);
      wmma3(a1, kk + 4);
      wmma3(a2, kk + 8);
      wmma3(a3, kk + 12);
    }

    __syncthreads();                 // everyone done reading sW[cur]
#if HAVE_TDM
    if (c + 1 < H_ / 128 && wave == 0)
      __builtin_amdgcn_s_wait_tensorcnt(0);   // next buffer landed
#endif
    __syncthreads();                 // next buffer visible to all waves
  }

  // D layout: VGPR v holds M = row0 + v + 8*lh, N = lm (+16 per N-tile)
  const float bb0 = bias[lm];
  const float bb1 = bias[16 + lm];
  const float bb2 = bias[32 + lm];
  #pragma unroll
  for (int v = 0; v < 8; ++v) {
    const int row = row0 + v + 8 * lh;
    float* out = feat + (size_t)row * T_;
    out[lm]      = acc0[v] + bb0;
    out[16 + lm] = acc1[v] + bb1;
    out[32 + lm] = acc2[v] + bb2;
  }
}

// ---------------------------------------------------------------------------
// Kernel 2: per-batch CRF: forward logsumexp (logZ), Viterbi + backtrack,
// gold score. One block per batch element; lane j owns tag j.
// Backpointers kept entirely in LDS (512*48 bytes).
// ---------------------------------------------------------------------------
__global__ __launch_bounds__(64)
void crf_kernel(const float* __restrict__ feat,
                const int*   __restrict__ mask,
                const int*   __restrict__ labels,
                const float* __restrict__ trans,
                const float* __restrict__ startv,
                const float* __restrict__ endv,
                float* __restrict__ pred_out,   // B*S (tags as float)
                float* __restrict__ partial) {  // B   (gold - logZ)
  __shared__ float sTrans[T_ * T_];
  __shared__ float sAlpha[T_];
  __shared__ float sV[T_];
  __shared__ unsigned char sBp[S_ * T_];
  __shared__ float sRed[64];
  __shared__ float sRes[3];

  const int b   = blockIdx.x;
  const int tid = threadIdx.x;
  const int j   = tid;
  const float* __restrict__ fb = feat + (size_t)b * S_ * T_;
  const int*   __restrict__ mb = mask + b * S_;
  const int*   __restrict__ lb = labels + b * S_;

  for (int idx = tid; idx < T_ * T_; idx += 64) sTrans[idx] = trans[idx];
  if (tid < T_) {
    const float v0 = startv[j] + fb[j];
    sAlpha[j] = v0;
    sV[j] = v0;
  }
  __syncthreads();

  for (int s = 1; s < S_; ++s) {
    const int m = mb[s];
    float newA = 0.f, newV = 0.f;
    int bp = 0;
    if (tid < T_) {
      const float emit = fb[s * T_ + j];
      // logsumexp over i of alpha[i] + trans[i][j]
      float mx = -__builtin_inff();
      #pragma unroll 4
      for (int i = 0; i < T_; ++i)
        mx = fmaxf(mx, sAlpha[i] + sTrans[i * T_ + j]);
      float sum = 0.f;
      #pragma unroll 4
      for (int i = 0; i < T_; ++i)
        sum += __expf(sAlpha[i] + sTrans[i * T_ + j] - mx);
      newA = mx + __logf(sum) + emit;
      // Viterbi max/argmax (first-max wins, matches jnp.argmax)
      float best = -__builtin_inff();
      int bi = 0;
      #pragma unroll 4
      for (int i = 0; i < T_; ++i) {
        const float sc = sV[i] + sTrans[i * T_ + j];
        if (sc > best) { best = sc; bi = i; }
      }
      newV = best + emit;
      bp = bi;
    }
    __syncthreads();
    if (tid < T_) {
      if (m > 0) {
        sAlpha[j] = newA;
        sV[j]     = newV;
        sBp[s * T_ + j] = (unsigned char)bp;
      } else {
        sBp[s * T_ + j] = (unsigned char)j;   // identity backpointer
      }
    }
    __syncthreads();
  }

  // gold score partial sums (parallel over s)
  float es = 0.f, ts = 0.f;
  int cnt = 0;
  for (int s = tid; s < S_; s += 64) {
    const int m = mb[s];
    cnt += m;
    if (m > 0) {
      es += fb[s * T_ + lb[s]];
      if (s > 0) ts += sTrans[lb[s - 1] * T_ + lb[s]];
    }
  }
  sRed[tid] = es; __syncthreads();
  if (tid == 0) { float a = 0.f; for (int i = 0; i < 64; ++i) a += sRed[i]; sRes[0] = a; }
  __syncthreads();
  sRed[tid] = ts; __syncthreads();
  if (tid == 0) { float a = 0.f; for (int i = 0; i < 64; ++i) a += sRed[i]; sRes[1] = a; }
  __syncthreads();
  sRed[tid] = (float)cnt; __syncthreads();

  if (tid == 0) {
    float fcnt = 0.f;
    for (int i = 0; i < 64; ++i) fcnt += sRed[i];
    const int CNT = (int)(fcnt + 0.5f);

    // logZ = lse(alpha + end)
    float mx = -__builtin_inff();
    for (int t = 0; t < T_; ++t) mx = fmaxf(mx, sAlpha[t] + endv[t]);
    float sum = 0.f;
    for (int t = 0; t < T_; ++t) sum += __expf(sAlpha[t] + endv[t] - mx);
    const float logZ = mx + __logf(sum);

    // Viterbi terminal tag
    float best = -__builtin_inff();
    int last = 0;
    for (int t = 0; t < T_; ++t) {
      const float sc = sV[t] + endv[t];
      if (sc > best) { best = sc; last = t; }
    }
    // backtrack through LDS backpointers
    int tag = last;
    for (int s = S_ - 1; s >= 1; --s) {
      pred_out[b * S_ + s] = (float)(tag * mb[s]);
      tag = (int)sBp[s * T_ + tag];
    }
    pred_out[b * S_ + 0] = (float)(tag * mb[0]);

    const float gold = startv[lb[0]] + sRes[0] + sRes[1] + endv[lb[CNT - 1]];
    partial[b] = gold - logZ;
  }
}

// ---------------------------------------------------------------------------
// Kernel 3: loss = -sum_b (gold[b] - logZ[b])
// ---------------------------------------------------------------------------
__global__ __launch_bounds__(64)
void loss_kernel(const float* __restrict__ partial, float* __restrict__ loss) {
  __shared__ float s[64];
  s[threadIdx.x] = partial[threadIdx.x];
  __syncthreads();
  if (threadIdx.x == 0) {
    float a = 0.f;
    for (int i = 0; i < 64; ++i) a += s[i];
    *loss = -a;
  }
}

extern "C" void kernel_launch(void* const* d_in, const int* in_sizes, int n_in,
                              void* d_out, int out_size, void* d_ws, size_t ws_size,
                              hipStream_t stream) {
  const float* x      = (const float*)d_in[0];
  const int*   mask   = (const int*)  d_in[1];
  const int*   labels = (const int*)  d_in[2];   // jax x64 off -> int32
  const float* W      = (const float*)d_in[3];
  const float* bias   = (const float*)d_in[4];
  const float* trans  = (const float*)d_in[5];
  const float* startv = (const float*)d_in[6];
  const float* endv   = (const float*)d_in[7];

  float* feat = (float*)d_out;                        // B*S*T
  float* pred = feat + (size_t)B_ * S_ * T_;          // B*S
  float* loss = pred + (size_t)B_ * S_;               // 1
  float* partial = (float*)d_ws;                      // B floats scratch

  // M tiles = 32768/16 = 2048; 8 waves (tiles) per block -> 256 blocks
  gemm_features_kernel<<<256, 256, 0, stream>>>(x, W, bias, feat);
  crf_kernel<<<B_, 64, 0, stream>>>(feat, mask, labels, trans, startv, endv, pred, partial);
  loss_kernel<<<1, 64, 0, stream>>>(partial, loss);
}